// DRCLoss_2121713844806
// MI455X (gfx1250) — compile-verified
//
#include <hip/hip_runtime.h>
#include <math.h>

typedef __attribute__((ext_vector_type(2))) float v2f;
typedef __attribute__((ext_vector_type(8))) float v8f;

#define NROWS 8192
#define KDIM  128
#define INV_T_AF 2.0f      // 1 / 0.5
#define LMBDA_   0.5f

// ws layout in floats (every slot fully overwritten each call -> deterministic,
// no init pass, no atomics):
#define WS_AFPART  0                        // 512 per-block af partial sums
#define WS_COLPART 512                      // 256 blocks x 128 col partials
#define WS_SIMPART (512 + 256*128)          // 8 segments x 128x128 partial sim_ap
#define WS_SIMAP   (WS_SIMPART + 8*128*128) // reduced 128x128 sim_ap

// ---------------------------------------------------------------------------
// Kernel 1: fused GEMM (af1 . af2^T) + online logsumexp over columns.
// Grid: 512 blocks (16 rows each), 128 threads = 4 waves.
// Each wave sweeps 128 of the 512 column tiles; fp32 WMMA 16x16x4, K=128.
// B chunks are batch-loaded into a register array (clauseable, no per-chunk
// wait), and the next tile is prefetched while the WMMA burst executes.
// ---------------------------------------------------------------------------
__global__ __launch_bounds__(128) void af_kernel(const float* __restrict__ af1,
                                                 const float* __restrict__ af2,
                                                 float* __restrict__ ws) {
    __shared__ float lds_m[4][16][16];
    __shared__ float lds_s[4][16][16];
    __shared__ float lds_d[16][8];
    __shared__ float contrib[16];

    const int tid  = threadIdx.x;
    const int w    = tid >> 5;
    const int lane = tid & 31;
    const int i0   = blockIdx.x * 16;

    const int m    = lane & 15;            // A: row M / B: col N
    const int koff = (lane >> 4) << 1;     // K pair select within 4-chunk

    // A tile 16x128 resident in 64 VGPRs (v2f per K-chunk), loaded once.
    v2f a[32];
    const float* arow = af1 + (i0 + m) * KDIM + koff;
#pragma unroll
    for (int c = 0; c < 32; ++c)
        a[c] = *(const v2f*)(arow + 4 * c);

    float run_m[8], run_s[8];
#pragma unroll
    for (int r = 0; r < 8; ++r) { run_m[r] = -3.0e38f; run_s[r] = 0.0f; }

    // Wave w handles column tiles jt = w, w+4, ..., w+508 (af2 is L2-resident).
    for (int t = 0; t < 128; ++t) {
        const int j0 = (w + 4 * t) * 16;
        const float* brow = af2 + (j0 + m) * KDIM + koff;

        // Batch-load the full 16x128 B tile slice for this lane: 32
        // independent b64 loads the compiler can clause.
        v2f b[32];
#pragma unroll
        for (int c = 0; c < 32; ++c)
            b[c] = *(const v2f*)(brow + 4 * c);

        // Prefetch next tile (4 x 128B cachelines per row; 16 rows covered
        // by the 32 lanes) while the WMMA burst below executes.
        if (t < 127) {
            const float* nbrow = af2 + ((w + 4 * (t + 1)) * 16 + m) * KDIM;
#pragma unroll
            for (int q = 0; q < 4; ++q)
                __builtin_prefetch((const void*)(nbrow + 32 * q), 0, 3);
        }

        v8f c0 = {};
        v8f c1 = {};
#pragma unroll
        for (int c = 0; c < 16; ++c) {
            c0 = __builtin_amdgcn_wmma_f32_16x16x4_f32(false, a[2 * c],     false, b[2 * c],
                                                       (short)0, c0, false, false);
            c1 = __builtin_amdgcn_wmma_f32_16x16x4_f32(false, a[2 * c + 1], false, b[2 * c + 1],
                                                       (short)0, c1, false, false);
        }
        // Online logsumexp update: each lane owns 8 (row, col=j0+m) entries.
#pragma unroll
        for (int r = 0; r < 8; ++r) {
            float v  = (c0[r] + c1[r]) * INV_T_AF;
            float mn = fmaxf(run_m[r], v);
            run_s[r] = run_s[r] * __expf(run_m[r] - mn) + __expf(v - mn);
            run_m[r] = mn;
        }
    }

    // Spill per-lane softmax state to LDS: [wave][row 0..15][col-lane 0..15]
#pragma unroll
    for (int r = 0; r < 8; ++r) {
        const int mm = r + ((lane >> 4) << 3);
        lds_m[w][mm][m] = run_m[r];
        lds_s[w][mm][m] = run_s[r];
    }

    // Diagonal dot products: 8 threads x 16 elements per row.
    {
        const int dm  = tid >> 3;   // 0..15
        const int seg = tid & 7;    // 0..7
        const float* x = af1 + (i0 + dm) * KDIM + seg * 16;
        const float* y = af2 + (i0 + dm) * KDIM + seg * 16;
        float s = 0.f;
#pragma unroll
        for (int k = 0; k < 16; ++k) s += x[k] * y[k];
        lds_d[dm][seg] = s;
    }
    __syncthreads();

    if (tid < 16) {
        float d = 0.f;
#pragma unroll
        for (int seg = 0; seg < 8; ++seg) d += lds_d[tid][seg];
        const float sim_ii = d * INV_T_AF;
        // merge 64 partial (max,sumexp) states for this row
        float M = -3.0e38f;
        for (int wv = 0; wv < 4; ++wv)
            for (int n = 0; n < 16; ++n)
                M = fmaxf(M, lds_m[wv][tid][n]);
        float S = 0.f;
        for (int wv = 0; wv < 4; ++wv)
            for (int n = 0; n < 16; ++n)
                S += lds_s[wv][tid][n] * __expf(lds_m[wv][tid][n] - M);
        contrib[tid] = (M + __logf(S)) - sim_ii;   // -log p[i,i]
    }
    __syncthreads();
    if (tid == 0) {
        float s = 0.f;
#pragma unroll
        for (int r = 0; r < 16; ++r) s += contrib[r];
        ws[WS_AFPART + blockIdx.x] = s;
    }
}

// ---------------------------------------------------------------------------
// Kernel 2: partial sim_ap = ap1^T . ap2  (128x128, contraction over N=8192
// split into 8 segments of 1024 rows -> 512 waves across 128 blocks).
// Each wave builds one partial 16x16 tile via fp32 WMMA; loads batched per
// 16-deep K step so they clause.
// ---------------------------------------------------------------------------
__global__ __launch_bounds__(128) void ap_gemm(const float* __restrict__ ap1,
                                               const float* __restrict__ ap2,
                                               float* __restrict__ ws) {
    const int tid  = threadIdx.x;
    const int w    = tid >> 5;
    const int lane = tid & 31;
    const int gw   = blockIdx.x * 4 + w;     // 0..511
    const int seg  = gw >> 6;                // 0..7 : contraction segment
    const int tile = gw & 63;                // 0..63 : output tile
    const int k0   = (tile >> 3) * 16;       // output row tile (k index)
    const int l0   = (tile & 7) * 16;        // output col tile (l index)
    const int m    = lane & 15;
    const int coff = (lane >> 4) << 1;       // contraction-pair select

    v8f c0 = {};
    v8f c1 = {};
    const float* A = ap1 + k0 + m;           // column-major walk over ap1
    const float* B = ap2 + l0 + m;
    const int nbeg = seg * 1024;
    const int nend = nbeg + 1024;
    for (int n0 = nbeg; n0 < nend; n0 += 16) {
        v2f av[4], bv[4];
#pragma unroll
        for (int q = 0; q < 4; ++q) {
            const int nb = n0 + 4 * q + coff;
            av[q].x = A[nb * KDIM];       av[q].y = A[(nb + 1) * KDIM];
            bv[q].x = B[nb * KDIM];       bv[q].y = B[(nb + 1) * KDIM];
        }
        c0 = __builtin_amdgcn_wmma_f32_16x16x4_f32(false, av[0], false, bv[0],
                                                   (short)0, c0, false, false);
        c1 = __builtin_amdgcn_wmma_f32_16x16x4_f32(false, av[1], false, bv[1],
                                                   (short)0, c1, false, false);
        c0 = __builtin_amdgcn_wmma_f32_16x16x4_f32(false, av[2], false, bv[2],
                                                   (short)0, c0, false, false);
        c1 = __builtin_amdgcn_wmma_f32_16x16x4_f32(false, av[3], false, bv[3],
                                                   (short)0, c1, false, false);
    }
    float* simp = ws + WS_SIMPART + seg * (KDIM * KDIM);  // T_AP == 1.0
#pragma unroll
    for (int r = 0; r < 8; ++r) {
        const int kk = k0 + r + ((lane >> 4) << 3);
        simp[kk * KDIM + l0 + m] = c0[r] + c1[r];
    }
}

// ---------------------------------------------------------------------------
// Kernel 2b: reduce the 8 partial sim_ap matrices into one.
// ---------------------------------------------------------------------------
__global__ __launch_bounds__(128) void ap_reduce(float* __restrict__ ws) {
    const int r = blockIdx.x;        // row 0..127
    const int l = threadIdx.x;       // col 0..127
    float s = 0.f;
#pragma unroll
    for (int seg = 0; seg < 8; ++seg)
        s += ws[WS_SIMPART + seg * (KDIM * KDIM) + r * KDIM + l];
    ws[WS_SIMAP + r * KDIM + l] = s;
}

// ---------------------------------------------------------------------------
// Kernel 3: column partial sums of ap1 (coalesced; no atomics).
// 256 blocks x 128 threads; block b sums rows [32b, 32b+32).
// ---------------------------------------------------------------------------
__global__ __launch_bounds__(128) void colsum_kernel(const float* __restrict__ ap1,
                                                     float* __restrict__ ws) {
    const int b = blockIdx.x;       // 0..255
    const int k = threadIdx.x;      // 0..127
    float s = 0.f;
#pragma unroll 8
    for (int r = 0; r < 32; ++r)
        s += ap1[(b * 32 + r) * KDIM + k];
    ws[WS_COLPART + b * 128 + k] = s;
}

// ---------------------------------------------------------------------------
// Kernel 4: final reductions -> (total, af_loss, ap_loss, reg_loss).
// ---------------------------------------------------------------------------
__global__ __launch_bounds__(128) void finalize(const float* __restrict__ ws,
                                                float* __restrict__ out) {
    __shared__ float r_ap[128];
    __shared__ float r_reg[128];
    __shared__ float r_af[128];
    const int t = threadIdx.x;
    const float* sim = ws + WS_SIMAP;

    // ap row t: logsumexp - diagonal
    float M = -3.0e38f;
    for (int l = 0; l < KDIM; ++l) M = fmaxf(M, sim[t * KDIM + l]);
    float S = 0.f;
    for (int l = 0; l < KDIM; ++l) S += __expf(sim[t * KDIM + l] - M);
    r_ap[t] = (M + __logf(S)) - sim[t * KDIM + t];

    // regularizer: colsum[t]^2 (sum of 256 partials)
    float cs = 0.f;
    for (int b = 0; b < 256; ++b) cs += ws[WS_COLPART + b * 128 + t];
    r_reg[t] = cs * cs;

    // af partials: 512 block sums, 4 per thread
    float af = 0.f;
#pragma unroll
    for (int i = 0; i < 4; ++i) af += ws[WS_AFPART + t + 128 * i];
    r_af[t] = af;

    __syncthreads();
    for (int off = 64; off > 0; off >>= 1) {
        if (t < off) {
            r_ap[t]  += r_ap[t + off];
            r_reg[t] += r_reg[t + off];
            r_af[t]  += r_af[t + off];
        }
        __syncthreads();
    }
    if (t == 0) {
        const float ap_loss = r_ap[0] / (float)KDIM;
        const float reg     = r_reg[0] / (float)NROWS;
        const float af_loss = r_af[0] / (float)NROWS;
        out[0] = af_loss + ap_loss + LMBDA_ * reg;
        out[1] = af_loss;
        out[2] = ap_loss;
        out[3] = reg;
    }
}

extern "C" void kernel_launch(void* const* d_in, const int* in_sizes, int n_in,
                              void* d_out, int out_size, void* d_ws, size_t ws_size,
                              hipStream_t stream) {
    const float* ap1 = (const float*)d_in[0];
    const float* af1 = (const float*)d_in[1];
    const float* ap2 = (const float*)d_in[2];
    const float* af2 = (const float*)d_in[3];
    float* ws  = (float*)d_ws;
    float* out = (float*)d_out;

    af_kernel<<<512, 128, 0, stream>>>(af1, af2, ws);
    ap_gemm<<<128, 128, 0, stream>>>(ap1, ap2, ws);
    ap_reduce<<<128, 128, 0, stream>>>(ws);
    colsum_kernel<<<256, 128, 0, stream>>>(ap1, ws);
    finalize<<<1, 128, 0, stream>>>(ws, out);
}